// cMLP_91182155694693
// MI455X (gfx1250) — compile-verified
//
#include <hip/hip_runtime.h>
#include <hip/hip_bf16.h>

typedef __attribute__((ext_vector_type(2))) float v2f;
typedef __attribute__((ext_vector_type(8))) float v8f;

// Problem constants (from reference)
#define NETS   64
#define BATCH  16
#define TLEN   2048
#define LAG    5
#define TOUT   2044          // TLEN - LAG + 1
#define H0     32
#define H1     32
#define K0     320           // NETS * LAG (layer-0 K)
#define KT0    80            // K0 / 4
#define KT1    8             // H0 / 4
#define KT2    8             // H1 / 4

#define TILES_PER_WAVE 4     // each wave processes 4 time-tiles (amortize LDS staging)

// Packed B-fragment region sizes (floats)
#define SZ0 (NETS * KT0 * 2 * 64)   // 655360
#define SZ1 (NETS * KT1 * 2 * 64)   // 65536
#define SZ2 (NETS * KT2 * 64)       // 32768

// ---------------------------------------------------------------------------
// Pack W0/W1/W2 into per-(K-tile, N-tile) 64-float B-fragment blocks.
// B-fragment layout (mirror of documented 16x4 f32 A layout, M<->N swapped):
//   lane L  (<16): reg0 = B[k0+0][N=L],  reg1 = B[k0+1][N=L]
//   lane L+16    : reg0 = B[k0+2][N=L],  reg1 = B[k0+3][N=L]
// Block element index s = lane*2 + reg.
// ---------------------------------------------------------------------------
__global__ void cmlp_pack_kernel(const float* __restrict__ W0,
                                 const float* __restrict__ W1,
                                 const float* __restrict__ W2,
                                 float* __restrict__ pack) {
  int e = blockIdx.x * blockDim.x + threadIdx.x;
  if (e < SZ0) {
    // Layer 0: B0[k][h] = W0[net][h][c][l] with k = l*64 + c  (matches
    // contiguous X window ordering).  W0 flat: ((net*32+h)*64+c)*5 + l
    int net = e / (KT0 * 2 * 64);
    int r0  = e % (KT0 * 2 * 64);
    int kt  = r0 / 128;
    int r1  = r0 % 128;
    int nt  = r1 / 64;
    int s   = r1 % 64;
    int lane = s >> 1, rr = s & 1;
    int k = kt * 4 + ((lane >= 16) ? 2 : 0) + rr;
    int h = nt * 16 + (lane & 15);
    int c = k & 63, l = k >> 6;
    pack[e] = W0[((net * 32 + h) * 64 + c) * 5 + l];
  } else if (e < SZ0 + SZ1) {
    // Layer 1: B1[i][o] = W1[net][o][i].  W1 flat: (net*32+o)*32 + i
    int e1  = e - SZ0;
    int net = e1 / (KT1 * 2 * 64);
    int r0  = e1 % (KT1 * 2 * 64);
    int kt  = r0 / 128;
    int r1  = r0 % 128;
    int nt  = r1 / 64;
    int s   = r1 % 64;
    int lane = s >> 1, rr = s & 1;
    int i = kt * 4 + ((lane >= 16) ? 2 : 0) + rr;
    int o = nt * 16 + (lane & 15);
    pack[e] = W1[(net * 32 + o) * 32 + i];
  } else if (e < SZ0 + SZ1 + SZ2) {
    // Layer 2: one-hot column B: B2[i][0] = W2[net][0][i], other cols 0.
    int e2  = e - SZ0 - SZ1;
    int net = e2 / (KT2 * 64);
    int r0  = e2 % (KT2 * 64);
    int kt  = r0 / 64;
    int s   = r0 % 64;
    int lane = s >> 1, rr = s & 1;
    int i   = kt * 4 + ((lane >= 16) ? 2 : 0) + rr;
    int col = lane & 15;
    pack[e] = (col == 0) ? W2[net * 32 + i] : 0.0f;
  }
}

// ---------------------------------------------------------------------------
// Main kernel: block = (net, batch, quadrant of 32 tiles); wave = 4 time
// tiles of 16 rows each. Per tile: layer0 80x2 WMMA f32 16x16x4, layer1 8x2,
// layer2 8 (one-hot column).
// ---------------------------------------------------------------------------
__global__ __launch_bounds__(256) void cmlp_main_kernel(
    const float* __restrict__ X,
    const float* __restrict__ b0,
    const float* __restrict__ b1,
    const float* __restrict__ b2,
    const float* __restrict__ pack,
    float* __restrict__ out) {

  __shared__ __align__(16) float sB0[KT0 * 2 * 64];   // 40 KB
  __shared__ __align__(16) float sB1[KT1 * 2 * 64];   // 4 KB
  __shared__ __align__(16) float sB2[KT2 * 64];       // 2 KB
  __shared__ float sBias0[32];
  __shared__ float sBias1[32];
  __shared__ float sBias2[1];
  __shared__ __align__(16) float sH[8][16 * 32];      // 16 KB per-wave scratch

  const int tid = threadIdx.x;
  const int bid = blockIdx.x;
  const int net = bid / (BATCH * 4);
  const int rem = bid % (BATCH * 4);
  const int bat = rem >> 2;
  const int tq  = rem & 3;          // quadrant: 32 tiles (512 rows) each

  // Stage this network's packed weights into LDS (reused by all 8 waves x 4 tiles).
  {
    const float4* p0 = (const float4*)(pack + (size_t)net * (KT0 * 2 * 64));
    float4* s0 = (float4*)sB0;
    for (int i = tid; i < (KT0 * 2 * 64) / 4; i += 256) s0[i] = p0[i];
    const float* p1 = pack + (size_t)SZ0 + (size_t)net * (KT1 * 2 * 64);
    for (int i = tid; i < KT1 * 2 * 64; i += 256) sB1[i] = p1[i];
    const float* p2 = pack + (size_t)SZ0 + (size_t)SZ1 + (size_t)net * (KT2 * 64);
    for (int i = tid; i < KT2 * 64; i += 256) sB2[i] = p2[i];
    if (tid < 32) { sBias0[tid] = b0[net * 32 + tid]; sBias1[tid] = b1[net * 32 + tid]; }
    if (tid == 0) sBias2[0] = b2[net];
  }
  __syncthreads();

  const int wave  = tid >> 5;
  const int lane  = tid & 31;
  const int laneM = lane & 15;
  const int kofs  = (lane >> 4) << 1;   // 0 (lanes 0-15) or 2 (lanes 16-31)
  const int mhi   = (lane >> 4) * 8;    // C layout: lanes16-31 hold M=8..15

  // Per-block constants hoisted out of the tile loop.
  const float bias0_lo = sBias0[laneM];
  const float bias0_hi = sBias0[16 + laneM];
  const float bias1_lo = sBias1[laneM];
  const float bias1_hi = sBias1[16 + laneM];
  const float bias2_v  = sBias2[0];
  const float* Xbat = X + (size_t)bat * TLEN * 64;
  float* hs = sH[wave];

  for (int it = 0; it < TILES_PER_WAVE; ++it) {
    const int t0 = (tq * 32 + it * 8 + wave) * 16;

    // A-fragment row for this lane (clamped for the partial last tiles;
    // duplicate rows are computed but their stores are masked).
    int trow = t0 + laneM;
    if (trow > TOUT - 1) trow = TOUT - 1;
    const float* Arow = Xbat + (size_t)trow * 64;  // 320 contiguous f32 window

    // -------- Layer 0: [16 x 320] @ [320 x 32], bias in C --------
    v8f acc0, acc1;
    for (int r = 0; r < 8; ++r) { acc0[r] = bias0_lo; acc1[r] = bias0_hi; }
#pragma unroll 4
    for (int kt = 0; kt < KT0; ++kt) {
      v2f a   = *(const v2f*)(Arow + kt * 4 + kofs);
      v2f bb0 = *(const v2f*)(&sB0[(kt * 2 + 0) * 64 + lane * 2]);
      v2f bb1 = *(const v2f*)(&sB0[(kt * 2 + 1) * 64 + lane * 2]);
      acc0 = __builtin_amdgcn_wmma_f32_16x16x4_f32(false, a, false, bb0,
                                                   (short)0, acc0, false, false);
      acc1 = __builtin_amdgcn_wmma_f32_16x16x4_f32(false, a, false, bb1,
                                                   (short)0, acc1, false, false);
    }

    // ReLU + transpose C-layout -> row-major [m][i] via per-wave LDS scratch.
    for (int r = 0; r < 8; ++r) {
      const int m = mhi + r;
      hs[m * 32 + laneM]      = fmaxf(acc0[r], 0.0f);
      hs[m * 32 + 16 + laneM] = fmaxf(acc1[r], 0.0f);
    }
    __syncthreads();

    // -------- Layer 1: [16 x 32] @ [32 x 32], bias in C --------
    v8f d0, d1;
    for (int r = 0; r < 8; ++r) { d0[r] = bias1_lo; d1[r] = bias1_hi; }
#pragma unroll
    for (int kt = 0; kt < KT1; ++kt) {
      v2f a   = *(const v2f*)(&hs[laneM * 32 + kt * 4 + kofs]);
      v2f bb0 = *(const v2f*)(&sB1[(kt * 2 + 0) * 64 + lane * 2]);
      v2f bb1 = *(const v2f*)(&sB1[(kt * 2 + 1) * 64 + lane * 2]);
      d0 = __builtin_amdgcn_wmma_f32_16x16x4_f32(false, a, false, bb0,
                                                 (short)0, d0, false, false);
      d1 = __builtin_amdgcn_wmma_f32_16x16x4_f32(false, a, false, bb1,
                                                 (short)0, d1, false, false);
    }
    __syncthreads();

    // ReLU + write h1 back to scratch (overwrites h0).
    for (int r = 0; r < 8; ++r) {
      const int m = mhi + r;
      hs[m * 32 + laneM]      = fmaxf(d0[r], 0.0f);
      hs[m * 32 + 16 + laneM] = fmaxf(d1[r], 0.0f);
    }
    __syncthreads();

    // -------- Layer 2: [16 x 32] @ [32 x 1] via one-hot-column WMMA --------
    v8f d2;
    for (int r = 0; r < 8; ++r) d2[r] = bias2_v;
#pragma unroll
    for (int kt = 0; kt < KT2; ++kt) {
      v2f a  = *(const v2f*)(&hs[laneM * 32 + kt * 4 + kofs]);
      v2f bb = *(const v2f*)(&sB2[kt * 64 + lane * 2]);
      d2 = __builtin_amdgcn_wmma_f32_16x16x4_f32(false, a, false, bb,
                                                 (short)0, d2, false, false);
    }

    // Column N=0 of D holds the 16 outputs (lanes 0 and 16).
    if (laneM == 0) {
      for (int r = 0; r < 8; ++r) {
        const int t = t0 + mhi + r;
        if (t < TOUT)
          out[((size_t)bat * TOUT + t) * 64 + net] = d2[r];
      }
    }
    __syncthreads();   // protect sH reuse across tile iterations
  }
}

// ---------------------------------------------------------------------------
extern "C" void kernel_launch(void* const* d_in, const int* in_sizes, int n_in,
                              void* d_out, int out_size, void* d_ws, size_t ws_size,
                              hipStream_t stream) {
  const float* X  = (const float*)d_in[0];
  const float* W0 = (const float*)d_in[1];
  const float* b0 = (const float*)d_in[2];
  const float* W1 = (const float*)d_in[3];
  const float* b1 = (const float*)d_in[4];
  const float* W2 = (const float*)d_in[5];
  const float* b2 = (const float*)d_in[6];
  float* out  = (float*)d_out;
  float* pack = (float*)d_ws;   // needs (SZ0+SZ1+SZ2)*4 ~= 2.95 MB

  const int packTotal = SZ0 + SZ1 + SZ2;
  cmlp_pack_kernel<<<(packTotal + 255) / 256, 256, 0, stream>>>(W0, W1, W2, pack);
  cmlp_main_kernel<<<NETS * BATCH * 4, 256, 0, stream>>>(X, b0, b1, b2, pack, out);
}